// RoutingLayer_1546188227171
// MI455X (gfx1250) — compile-verified
//
#include <hip/hip_runtime.h>

typedef __attribute__((ext_vector_type(2))) float v2f;
typedef __attribute__((ext_vector_type(8))) float v8f;
typedef int vs4i __attribute__((vector_size(16)));   // matches builtin param type

static constexpr int KCAP  = 8;    // capsules
static constexpr int CDD   = 16;   // dims per capsule
static constexpr int DFULL = 128;  // K*DD
static constexpr int MNB   = 16;   // neighbors
static constexpr int ZPAD  = 132;  // LDS row stride (floats) to break bank conflicts

// ---------------------------------------------------------------------------
// CDNA5 async global->LDS copy (GLOBAL_LOAD_ASYNC_TO_LDS_B128, ASYNCcnt).
// Falls back to a synchronous float4 copy if the builtin is unavailable.
// ---------------------------------------------------------------------------
#if __has_builtin(__builtin_amdgcn_global_load_async_to_lds_b128)
#define HAS_ASYNC_LDS 1
#else
#define HAS_ASYNC_LDS 0
#endif

__device__ __forceinline__ void async_ld_b128(const float* gsrc, float* ldst) {
#if HAS_ASYNC_LDS
  __builtin_amdgcn_global_load_async_to_lds_b128(
      (__attribute__((address_space(1))) vs4i*)(uintptr_t)gsrc,
      (__attribute__((address_space(3))) vs4i*)(uint32_t)(uintptr_t)ldst,
      0, 0);
#else
  *(float4*)ldst = *(const float4*)gsrc;
#endif
}

__device__ __forceinline__ void wait_async_lds() {
#if HAS_ASYNC_LDS
#if __has_builtin(__builtin_amdgcn_s_wait_asynccnt)
  __builtin_amdgcn_s_wait_asynccnt(0);
#else
  asm volatile("s_wait_asynccnt 0x0" ::: "memory");
#endif
#endif
}

// ---------------------------------------------------------------------------
// Kernel 1: firstLayer. One wave32 per 16-node tile.
// Per-capsule 16(node)x16(e)x16(d) fp32 GEMM via 4 chained V_WMMA_F32_16X16X4_F32.
// ---------------------------------------------------------------------------
__global__ __launch_bounds__(32) void caps_first_layer(
    const float* __restrict__ x,
    const float* __restrict__ fc_w,   // (K, DD, DD) [k][d][e]
    const float* __restrict__ fc_b,   // (K, DD)
    float* __restrict__ z_full,       // (n+1, 128)
    int n_nodes)
{
  const int lane  = (int)threadIdx.x;
  const int node0 = (int)blockIdx.x * 16;

  if (node0 >= n_nodes) {             // exactly one pad block: zero row N
    for (int i = lane; i < DFULL; i += 32)
      z_full[(size_t)n_nodes * DFULL + i] = 0.0f;
    return;
  }

  __shared__ float xs[16][ZPAD];
  __shared__ float ys[16][CDD + 1];
  __shared__ float invn[16];

  // load 16x128 tile: 2 lanes per row, 16x b128 async copies per lane
  {
    const int r  = lane >> 1;
    const int c0 = (lane & 1) * 64;
    if (node0 + r < n_nodes) {
      const float* src = x + (size_t)(node0 + r) * DFULL + c0;
      float* dst = &xs[r][c0];        // (r*132+c0)*4 bytes is 16B aligned
      #pragma unroll
      for (int t = 0; t < 16; ++t) async_ld_b128(src + 4 * t, dst + 4 * t);
    } else {                          // ragged rows: zero-fill
      const float4 zero = make_float4(0.f, 0.f, 0.f, 0.f);
      #pragma unroll
      for (int t = 0; t < 16; ++t) *(float4*)&xs[r][c0 + 4 * t] = zero;
    }
  }
  wait_async_lds();
  __syncthreads();

  // per-(node,capsule) L2 normalize in place
  for (int e = lane; e < 16 * KCAP; e += 32) {
    const int r = e >> 3, k = e & 7;
    float* p = &xs[r][k * CDD];
    float s = 0.f;
    #pragma unroll
    for (int d = 0; d < CDD; ++d) s += p[d] * p[d];
    const float iv = 1.0f / fmaxf(sqrtf(s), 1e-12f);
    #pragma unroll
    for (int d = 0; d < CDD; ++d) p[d] *= iv;
  }
  __syncthreads();

  const int half = lane >> 4;
  const int lid  = lane & 15;

  for (int k = 0; k < KCAP; ++k) {
    v8f c = {};
    #pragma unroll
    for (int t = 0; t < 4; ++t) {     // K=16 as 4 chained K=4 fp32 WMMAs
      const int kb = 4 * t + 2 * half;
      v2f a, b;
      // A 16x4 fp32: lanes 0-15 M=lane K={4t,4t+1}; lanes 16-31 M=lane-16 K={4t+2,4t+3}
      a.x = xs[lid][k * CDD + kb + 0];
      a.y = xs[lid][k * CDD + kb + 1];
      // B 4x16 fp32: row = d (K), col = e (N), same half-lane K split
      b.x = fc_w[(k * CDD + kb + 0) * CDD + lid];
      b.y = fc_w[(k * CDD + kb + 1) * CDD + lid];
      c = __builtin_amdgcn_wmma_f32_16x16x4_f32(false, a, false, b,
                                                (short)0, c, false, false);
    }
    // D layout: VGPR v -> row M = v + 8*half, col N = lid
    const float bias = fc_b[k * CDD + lid];
    #pragma unroll
    for (int v = 0; v < 8; ++v)
      ys[v + 8 * half][lid] = fmaxf(c[v] + bias, 0.0f);
    __syncthreads();
    if (lane < 16) {                  // per-(node,k) norm over e
      float s = 0.f;
      #pragma unroll
      for (int e = 0; e < CDD; ++e) s += ys[lane][e] * ys[lane][e];
      invn[lane] = 1.0f / fmaxf(sqrtf(s), 1e-12f);
    }
    __syncthreads();
    #pragma unroll
    for (int v = 0; v < 8; ++v) {
      const int r = v + 8 * half;
      if (node0 + r < n_nodes)
        z_full[(size_t)(node0 + r) * DFULL + k * CDD + lid] = ys[r][lid] * invn[r];
    }
    __syncthreads();                  // WAR on ys before next capsule
  }
}

// ---------------------------------------------------------------------------
// Kernel 2: routing. One wave32 per node (single-wave WG -> barriers are NOPs).
// Neighbor gather uses the CDNA5 async global->LDS path (no VGPR staging).
// ---------------------------------------------------------------------------
__global__ __launch_bounds__(32) void caps_routing(
    const float* __restrict__ z_full,
    const int* __restrict__ neighbors,
    float* __restrict__ out_u,        // (n,128)
    float* __restrict__ out_p1,       // (n,16,8)
    const float* __restrict__ param_p,
    const int* __restrict__ max_iter_p,
    int n_nodes)
{
  const int n = (int)blockIdx.x;
  if (n >= n_nodes) return;
  const int lane = (int)threadIdx.x;

  const float pm  = param_p[0];
  const float pm1 = 1.0f - pm;
  const int   max_iter = max_iter_p[0];

  __shared__ float z[MNB][ZPAD];
  __shared__ float ub[DFULL];         // u_before (unnormalized accumulator)
  __shared__ float un[DFULL];         // normalized u for agreement logits
  __shared__ float pp[MNB][KCAP];
  __shared__ float p1s[MNB][KCAP];
  __shared__ float p2s[MNB][KCAP];
  __shared__ float invn[KCAP];
  __shared__ int   nb[MNB];

  if (lane < MNB) nb[lane] = neighbors[(size_t)n * MNB + lane];
  for (int i = lane; i < DFULL; i += 32) {
    const float v = z_full[(size_t)n * DFULL + i];
    ub[i] = v; un[i] = v;
  }
  __syncthreads();                    // nb[] visible to all lanes

  {                                   // gather: 2 lanes per neighbor row
    const int j  = lane >> 1;
    const int c0 = (lane & 1) * 64;
    const float* src = z_full + (size_t)nb[j] * DFULL + c0;
    float* dst = &z[j][c0];           // (j*132+c0)*4 bytes is 16B aligned
    #pragma unroll
    for (int t = 0; t < 16; ++t) async_ld_b128(src + 4 * t, dst + 4 * t);
  }
  wait_async_lds();
  __syncthreads();

  for (int it = 0; it < max_iter; ++it) {
    // agreement logits p[m,k] = <z[m,k,:], u[k,:]> (zero on iter 0)
    for (int e = lane; e < MNB * KCAP; e += 32) {
      const int m = e >> 3, k = e & 7;
      float acc = 0.f;
      if (it > 0) {
        const float* zr = &z[m][k * CDD];
        const float* ur = &un[k * CDD];
        #pragma unroll
        for (int d = 0; d < CDD; ++d) acc += zr[d] * ur[d];
      }
      pp[m][k] = acc;
    }
    __syncthreads();
    // dual softmax: lanes 0-7 over neighbors (p1), lanes 16-31 over capsules (p2)
    if (lane < KCAP) {
      const int k = lane;
      float mx = -3.0e38f;
      for (int m = 0; m < MNB; ++m) mx = fmaxf(mx, pp[m][k]);
      float ex[MNB], s = 0.f;
      for (int m = 0; m < MNB; ++m) { ex[m] = __expf(pp[m][k] - mx); s += ex[m]; }
      const float is = 1.0f / s;
      for (int m = 0; m < MNB; ++m) p1s[m][k] = ex[m] * is;
    } else if (lane >= 16) {
      const int m = lane - 16;
      float mx = -3.0e38f;
      for (int k = 0; k < KCAP; ++k) mx = fmaxf(mx, pp[m][k]);
      float ex[KCAP], s = 0.f;
      for (int k = 0; k < KCAP; ++k) { ex[k] = __expf(pp[m][k] - mx); s += ex[k]; }
      const float is = 1.0f / s;
      for (int k = 0; k < KCAP; ++k) p2s[m][k] = ex[k] * is;
    }
    __syncthreads();
    // u = param*u1 + (1-param)*u2 + u_before (fused weight per m)
    for (int e = lane; e < DFULL; e += 32) {
      const int k = e >> 4;
      float acc = ub[e];
      #pragma unroll
      for (int m = 0; m < MNB; ++m) {
        const float w = pm * p1s[m][k] + pm1 * p2s[m][k];
        acc += z[m][e] * w;           // z column index k*16+d == e
      }
      ub[e] = acc;
    }
    __syncthreads();
    if (it < max_iter - 1) {          // normalize except after last iteration
      if (lane < KCAP) {
        float s = 0.f;
        #pragma unroll
        for (int d = 0; d < CDD; ++d) { const float v = ub[lane * CDD + d]; s += v * v; }
        invn[lane] = 1.0f / fmaxf(sqrtf(s), 1e-12f);
      }
      __syncthreads();
      for (int e = lane; e < DFULL; e += 32) un[e] = ub[e] * invn[e >> 4];
      __syncthreads();
    }
  }

  // outputs: u (unnormalized) and final-iteration p1
  for (int e = lane; e < DFULL; e += 32) {
    out_u[(size_t)n * DFULL + e] = ub[e];
    out_p1[(size_t)n * (MNB * KCAP) + e] = p1s[e >> 3][e & 7];
  }
}

// ---------------------------------------------------------------------------
extern "C" void kernel_launch(void* const* d_in, const int* in_sizes, int n_in,
                              void* d_out, int out_size, void* d_ws, size_t ws_size,
                              hipStream_t stream) {
  const float* x         = (const float*)d_in[0];
  const int*   neighbors = (const int*)d_in[1];
  const float* fc_w      = (const float*)d_in[2];
  const float* fc_b      = (const float*)d_in[3];
  const int*   max_iter  = (const int*)d_in[4];
  const float* param     = (const float*)d_in[5];

  const int n = in_sizes[0] / DFULL;          // 50000

  float* z_full = (float*)d_ws;               // (n+1) x 128 fp32 = 25.6 MB (L2-resident)
  float* out_u  = (float*)d_out;
  float* out_p1 = out_u + (size_t)n * DFULL;

  const int tiles = (n + 15) / 16;
  caps_first_layer<<<tiles + 1, 32, 0, stream>>>(x, fc_w, fc_b, z_full, n);
  caps_routing<<<n, 32, 0, stream>>>(z_full, neighbors, out_u, out_p1,
                                     param, max_iter, n);
}